// StructureDecoder_39591008534761
// MI455X (gfx1250) — compile-verified
//
#include <hip/hip_runtime.h>

typedef float v2f __attribute__((ext_vector_type(2)));
typedef float v8f __attribute__((ext_vector_type(8)));

#define NN 16384
#define NE 524288
#define DD 64

// ---------------- degree / normalization ----------------

__global__ void k_deg_init(float* __restrict__ deg) {
    int i = blockIdx.x * 256 + threadIdx.x;
    deg[i] = 1.0f;   // self-loop contribution
}

__global__ void k_deg_count(const long long* __restrict__ col, float* __restrict__ deg) {
    int e = blockIdx.x * 256 + threadIdx.x;
    unsafeAtomicAdd(&deg[(int)col[e]], 1.0f);
}

__global__ void k_dinv(float* __restrict__ deg) {
    int i = blockIdx.x * 256 + threadIdx.x;
    deg[i] = rsqrtf(deg[i]);   // deg >= 1 always (self-loops)
}

// ---------------- h = z @ W  (WMMA f32 16x16x4) ----------------

__global__ void __launch_bounds__(256) k_linear(const float* __restrict__ z,
                                                const float* __restrict__ W,
                                                float* __restrict__ h) {
    __shared__ float Wl[64 * 68];   // stride-68 padded
    int t = threadIdx.x;
#pragma unroll
    for (int i = 0; i < 4; ++i) {
        int idx = t + i * 256;          // 1024 float4 = 64x64
        int r = idx >> 4;
        int c = (idx & 15) << 2;
        *(float4*)(&Wl[r * 68 + c]) = *(const float4*)(W + r * 64 + c);
    }
    __syncthreads();

    int wave = t >> 5, lane = t & 31;
    int tile = blockIdx.x * 8 + wave;   // 4096 tiles total (1024 x 4)
    int tm = tile >> 2;                 // row tile (16 rows)
    int tn = tile & 3;                  // col tile (16 cols)
    int mrow  = tm * 16 + (lane & 15);
    int ncol  = tn * 16 + (lane & 15);
    int khalf = (lane >> 4) << 1;

    v8f acc = {};
#pragma unroll
    for (int kk = 0; kk < 64; kk += 4) {
        int k = kk + khalf;
        v2f a = *(const v2f*)(z + mrow * 64 + k);       // A: z[16x4 tile]
        v2f b;
        b.x = Wl[k * 68 + ncol];                        // B: W[4x16 tile]
        b.y = Wl[(k + 1) * 68 + ncol];
        acc = __builtin_amdgcn_wmma_f32_16x16x4_f32(false, a, false, b,
                                                    (short)0, acc, false, false);
    }

    int mbase = tm * 16 + ((lane >> 4) << 3);
#pragma unroll
    for (int r = 0; r < 8; ++r)
        h[(mbase + r) * 64 + ncol] = acc[r];
}

// ---------------- self-loop init + scatter + bias/relu ----------------

__global__ void k_selfloop(const float* __restrict__ h, const float* __restrict__ dinv,
                           float* __restrict__ hagg) {
    int tid = blockIdx.x * 256 + threadIdx.x;
    int i = tid >> 6;
    float di = dinv[i];
    hagg[tid] = h[tid] * di * di;       // also initializes the accumulator
}

__global__ void k_scatter(const float* __restrict__ h,
                          const long long* __restrict__ row,
                          const long long* __restrict__ col,
                          const float* __restrict__ dinv,
                          float* __restrict__ hagg) {
    int tid = blockIdx.x * 256 + threadIdx.x;
    int e = tid >> 6;
    int d = tid & 63;
    int r = (int)row[e];
    int c = (int)col[e];
    float norm = dinv[r] * dinv[c];
    unsafeAtomicAdd(&hagg[c * 64 + d], h[r * 64 + d] * norm);
}

__global__ void k_bias_relu(float* __restrict__ hagg, const float* __restrict__ b) {
    int tid = blockIdx.x * 256 + threadIdx.x;
    float v = hagg[tid] + b[tid & 63];
    hagg[tid] = v > 0.0f ? v : 0.0f;
}

// ---------------- A_hat = h @ h^T  (WMMA f32, 128x128 tiles) ----------------

__global__ void __launch_bounds__(256) k_gram(const float* __restrict__ h,
                                              float* __restrict__ out) {
    __shared__ float Ab[128 * 68];
    __shared__ float Bb[128 * 68];
    int t = threadIdx.x;
    int row0 = blockIdx.y * 128;
    int col0 = blockIdx.x * 128;

#pragma unroll
    for (int i = 0; i < 8; ++i) {
        int idx = t + i * 256;          // 2048 float4 = 128x64
        int r = idx >> 4;
        int c = (idx & 15) << 2;
        *(float4*)(&Ab[r * 68 + c]) = *(const float4*)(h + (row0 + r) * 64 + c);
        *(float4*)(&Bb[r * 68 + c]) = *(const float4*)(h + (col0 + r) * 64 + c);
    }
    __syncthreads();

    int wave = t >> 5, lane = t & 31;
    int ra    = wave * 16 + (lane & 15);   // A fragment row in LDS
    int rb0   = lane & 15;                 // B fragment row within col tile
    int khalf = (lane >> 4) << 1;

    v8f acc[8] = {};
#pragma unroll
    for (int kk = 0; kk < 64; kk += 4) {
        int k = kk + khalf;
        v2f a = *(const v2f*)(&Ab[ra * 68 + k]);
#pragma unroll
        for (int j = 0; j < 8; ++j) {
            v2f b = *(const v2f*)(&Bb[(j * 16 + rb0) * 68 + k]);
            acc[j] = __builtin_amdgcn_wmma_f32_16x16x4_f32(false, a, false, b,
                                                           (short)0, acc[j], false, false);
        }
    }

    int n_l = lane & 15;
    int m_l = (lane >> 4) << 3;
    size_t mbase = (size_t)(row0 + wave * 16 + m_l);
#pragma unroll
    for (int j = 0; j < 8; ++j) {
        int gc = col0 + j * 16 + n_l;
#pragma unroll
        for (int r = 0; r < 8; ++r)
            out[(mbase + r) * (size_t)NN + gc] = acc[j][r];
    }
}

// ---------------- launch ----------------

extern "C" void kernel_launch(void* const* d_in, const int* in_sizes, int n_in,
                              void* d_out, int out_size, void* d_ws, size_t ws_size,
                              hipStream_t stream) {
    const float*     z  = (const float*)d_in[0];
    const long long* ei = (const long long*)d_in[1];   // int64 [2, E]
    const float*     W  = (const float*)d_in[2];
    const float*     b  = (const float*)d_in[3];
    float* out = (float*)d_out;

    float* ws   = (float*)d_ws;
    float* deg  = ws;                   // N floats, becomes dinv in place
    float* h    = ws + NN;              // N*64
    float* hagg = h + (size_t)NN * DD;  // N*64

    const long long* row = ei;
    const long long* col = ei + NE;

    k_deg_init <<<NN / 256, 256, 0, stream>>>(deg);
    k_deg_count<<<NE / 256, 256, 0, stream>>>(col, deg);
    k_dinv     <<<NN / 256, 256, 0, stream>>>(deg);

    k_linear   <<<512, 256, 0, stream>>>(z, W, h);                 // 4096 tiles / 8 waves

    k_selfloop <<<(NN * DD) / 256, 256, 0, stream>>>(h, deg, hagg);
    k_scatter  <<<(NE * DD) / 256, 256, 0, stream>>>(h, row, col, deg, hagg);
    k_bias_relu<<<(NN * DD) / 256, 256, 0, stream>>>(hagg, b);

    dim3 grid(NN / 128, NN / 128);      // 128 x 128 workgroups
    k_gram     <<<grid, 256, 0, stream>>>(hagg, out);
}